// HDMNet_52484500357140
// MI455X (gfx1250) — compile-verified
//
#include <hip/hip_runtime.h>
#include <hip/hip_bf16.h>
#include <math.h>

// Problem constants (from reference)
#define BATCH   8192
#define DIM     93     // input SPD size
#define DIM1    30     // BiMap output size
#define CLASSES 117
#define PK      96     // DIM padded to multiple of 16
#define PN      32     // DIM1 padded to multiple of 16
#define VDIM    900    // DIM1*DIM1

// LDS pitches produced by TDM pad_enable (1 DWORD pad per 32 stored DWORDs):
//   X tile : 96 data DW/row + 3 pads -> pitch 99 (odd => conflict-free columns)
//   W tile : 32 data DW/row + 1 pad  -> pitch 33
#define XPITCH  99
#define WPITCH  33

typedef __attribute__((ext_vector_type(2))) float v2f;
typedef __attribute__((ext_vector_type(8))) float v8f;
typedef __attribute__((ext_vector_type(4))) unsigned int v4u;
typedef __attribute__((ext_vector_type(8))) int v8i;
typedef __attribute__((ext_vector_type(4))) int v4i;

#if __has_builtin(__builtin_amdgcn_tensor_load_to_lds)
#define USE_TDM 1
#else
#define USE_TDM 0
#endif

#if USE_TDM
// ---------------------------------------------------------------------------
// Issue a 2-D TDM load: tile (tile_d0 x tile_d1) from a (tensor_d0 x tensor_d1)
// row-major f32 tensor into LDS at lds_addr. OOB reads return zero (gives us
// the 93->96 zero padding for free). pad: 1 DWORD per 32 stored DWORDs.
// D# bit packing per CDNA5 ISA 8.3/8.4 (group0 128b, group1 256b; groups 2/3
// zero for <=2D tiles). LDS offset assumption: low 32 bits of a flat pointer
// to __shared__ equal the wave-relative LDS byte address (ISA 10.2 aperture).
// This toolchain's builtin takes 6 args: (g0, g1, g2, g3, g_extra, cpol).
// ---------------------------------------------------------------------------
__device__ __forceinline__ void tdm_load_2d_f32(
    unsigned int lds_addr, const void* gptr,
    unsigned int tensor_d0, unsigned int tensor_d1,
    unsigned int tile_d0, unsigned int tile_d1,
    unsigned int stride0)
{
  unsigned long long ga = (unsigned long long)(uintptr_t)gptr;
  v4u g0;
  g0.x = 1u;                                   // count=1 valid descriptor
  g0.y = lds_addr;                             // LDS byte address
  g0.z = (unsigned int)ga;                     // global_addr[31:0]
  g0.w = (unsigned int)((ga >> 32) & 0x01FFFFFFu) | (2u << 30); // addr[56:32]|type=2
  v8i g1;
  // data_size=2 (4B) | pad_enable | pad_interval code 4 (32 DW) | pad_amount code 0 (1 DW)
  g1[0] = (int)((2u << 16) | (1u << 20) | (4u << 22));
  g1[1] = (int)((tensor_d0 & 0xFFFFu) << 16);                    // tensor_dim0[15:0]
  g1[2] = (int)((tensor_d0 >> 16) | ((tensor_d1 & 0xFFFFu) << 16));
  g1[3] = (int)((tensor_d1 >> 16) | (tile_d0 << 16));            // tile_dim0
  g1[4] = (int)(tile_d1);                                        // tile_dim1, tile_dim2=0
  g1[5] = (int)stride0;                                          // tensor_dim0_stride lo32
  g1[6] = 0;                                                     // stride0 hi16 | stride1 lo16
  g1[7] = 0;                                                     // stride1 hi32
  v4i z4 = {0, 0, 0, 0};
  v8i z8 = {0, 0, 0, 0, 0, 0, 0, 0};
  __builtin_amdgcn_tensor_load_to_lds(g0, g1, z4, z4, z8, 0);
}
#endif

// ---------------------------------------------------------------------------
// Kernel 1: BiMap  y = W^T X W  via two fp32 WMMA GEMM stages per batch.
//   Stage A: T = X * W      (96x96)*(96x32) -> (96x32)   [zero padded]
//   Stage B: y = W^T * T    (32x96)*(96x32) -> (32x32), store top-left 30x30
// One workgroup (256 thr = 8 waves) per batch. X and W staged in LDS by the
// Tensor Data Mover (wave 0 issues, s_wait_tensorcnt, barrier).
// Fragment layouts per CDNA5 ISA 7.12.2 (f32 16x16x4):
//   A 16x4: lanes 0-15 hold row M=lane, v0=K0 v1=K1; lanes16-31: v0=K2 v1=K3
//   B 4x16: lanes hold N=lane&15;       half 0: v0=K0 v1=K1; half 1: K2/K3
//   C/D:    vgpr r -> M = r + 8*(lane>=16), N = lane&15
// ---------------------------------------------------------------------------
__global__ __launch_bounds__(256) void bimap_kernel(
    const float* __restrict__ x, const float* __restrict__ Wg,
    float* __restrict__ yws)
{
  __shared__ float Xs[PK * XPITCH];   // 96x96 tile, TDM-padded pitch 99
  __shared__ float Ws[PK * WPITCH];   // 96x32 tile, TDM-padded pitch 33
  __shared__ float Ts[PK * WPITCH];   // T = X*W, same pitch-33 layout

  const int b   = blockIdx.x;
  const int tid = threadIdx.x;
  const float* Xb = x + (size_t)b * DIM * DIM;

#if USE_TDM
  if (tid < 32) {  // wave 0 issues both DMA descriptors (EXEC ignored by TDM)
    tdm_load_2d_f32((unsigned int)(uintptr_t)Xs, Xb,  DIM,  DIM, PK, PK, DIM);
    tdm_load_2d_f32((unsigned int)(uintptr_t)Ws, Wg, DIM1,  DIM, PN, PK, DIM1);
    __builtin_amdgcn_s_wait_tensorcnt((short)0);
  }
#else
  for (int idx = tid; idx < PK * PK; idx += 256) {
    int r = idx / PK, c = idx % PK;
    Xs[r * XPITCH + c + (c >> 5)] = (r < DIM && c < DIM) ? Xb[r * DIM + c] : 0.0f;
  }
  for (int idx = tid; idx < PK * PN; idx += 256) {
    int r = idx >> 5, c = idx & 31;
    Ws[r * WPITCH + c] = (r < DIM && c < DIM1) ? Wg[r * DIM1 + c] : 0.0f;
  }
#endif
  __syncthreads();

  const int w    = tid >> 5;
  const int lane = tid & 31;
  const int half = lane >> 4;
  const int l    = lane & 15;

  // Stage A: T = X * W ; 6x2 = 12 output tiles over 8 waves
  for (int t = w; t < 12; t += 8) {
    const int mt = t >> 1, nt = t & 1;
    const int arow = mt * 16 + l;
    const int ncol = nt * 16 + l;
    v8f acc = {};
    for (int k4 = 0; k4 < PK / 4; ++k4) {
      const int kb = k4 * 4 + half * 2;
      const int co = k4 >> 3;          // X column pad adjust: +1 DW per 32
      v2f a, bb;
      a.x  = Xs[arow * XPITCH + kb + co];
      a.y  = Xs[arow * XPITCH + kb + 1 + co];
      bb.x = Ws[kb * WPITCH + ncol];
      bb.y = Ws[(kb + 1) * WPITCH + ncol];
      acc = __builtin_amdgcn_wmma_f32_16x16x4_f32(
          false, a, false, bb, (short)0, acc, false, false);
    }
    for (int r = 0; r < 8; ++r)
      Ts[(mt * 16 + r + half * 8) * WPITCH + nt * 16 + l] = acc[r];
  }
  __syncthreads();

  // Stage B: y = W^T * T ; 2x2 tiles on waves 0..3 ; A[m][k] = W[k][m]
  if (w < 4) {
    const int mt = w >> 1, nt = w & 1;
    const int mrow = mt * 16 + l;
    const int ncol = nt * 16 + l;
    v8f acc = {};
    for (int k4 = 0; k4 < PK / 4; ++k4) {
      const int kb = k4 * 4 + half * 2;
      v2f a, bb;
      a.x  = Ws[kb * WPITCH + mrow];
      a.y  = Ws[(kb + 1) * WPITCH + mrow];
      bb.x = Ts[kb * WPITCH + ncol];
      bb.y = Ts[(kb + 1) * WPITCH + ncol];
      acc = __builtin_amdgcn_wmma_f32_16x16x4_f32(
          false, a, false, bb, (short)0, acc, false, false);
    }
    float* Yb = yws + (size_t)b * VDIM;
    for (int r = 0; r < 8; ++r) {
      const int row = mt * 16 + r + half * 8;
      const int col = nt * 16 + l;
      if (row < DIM1 && col < DIM1) Yb[row * DIM1 + col] = acc[r];
    }
  }
}

// ---------------------------------------------------------------------------
// Kernel 2: batched 30x30 symmetric eigendecomposition (cyclic Jacobi,
// fixed 10 sweeps -> deterministic), then logm = U diag(log d) U^T.
// One wave (32 lanes) per matrix; lane j owns column j. 8192 independent
// waves hide the serial rotation chain.
// ---------------------------------------------------------------------------
__global__ __launch_bounds__(32) void eig_log_kernel(
    const float* __restrict__ yws, float* __restrict__ logm)
{
  __shared__ float A[DIM1][DIM1 + 3];   // pitch 33: conflict-free column access
  __shared__ float U[DIM1][DIM1 + 3];
  __shared__ float ld[DIM1];

  const int b = blockIdx.x;
  const int j = threadIdx.x;
  const float* Y = yws + (size_t)b * VDIM;

  if (j < DIM1) {
    for (int i = 0; i < DIM1; ++i) {
      A[i][j] = Y[i * DIM1 + j];
      U[i][j] = (i == j) ? 1.0f : 0.0f;
    }
  }
  __syncthreads();

  for (int sweep = 0; sweep < 10; ++sweep) {
    for (int p = 0; p < DIM1 - 1; ++p) {
      for (int q = p + 1; q < DIM1; ++q) {
        // rotation parameters (uniform across lanes; LDS broadcast reads)
        const float apq = A[p][q];
        const float app = A[p][p];
        const float aqq = A[q][q];
        float c, s;
        if (fabsf(apq) > 1e-20f) {
          const float theta = (aqq - app) / (2.0f * apq);
          float t = 1.0f / (fabsf(theta) + sqrtf(theta * theta + 1.0f));
          if (theta < 0.0f) t = -t;
          c = 1.0f / sqrtf(t * t + 1.0f);
          s = t * c;
        } else {
          c = 1.0f; s = 0.0f;
        }
        // row rotation: A <- J^T A
        if (j < DIM1) {
          const float rp = A[p][j], rq = A[q][j];
          A[p][j] = c * rp - s * rq;
          A[q][j] = s * rp + c * rq;
        }
        __syncthreads();
        // column rotation: A <- A J ; eigenvector update U <- U J
        if (j < DIM1) {
          const float cp = A[j][p], cq = A[j][q];
          A[j][p] = c * cp - s * cq;
          A[j][q] = s * cp + c * cq;
          const float up = U[j][p], uq = U[j][q];
          U[j][p] = c * up - s * uq;
          U[j][q] = s * up + c * uq;
        }
        __syncthreads();
      }
    }
  }

  // log-eigenvalues (eigenvalues >= 1e-3 analytically; clamp for safety)
  if (j < DIM1) ld[j] = logf(fmaxf(A[j][j], 1e-30f));
  __syncthreads();

  // logm[i][j] = sum_p U[i][p] * ld[p] * U[j][p]; lane j computes column j
  if (j < DIM1) {
    float vj[DIM1];
    for (int p = 0; p < DIM1; ++p) vj[p] = U[j][p] * ld[p];
    float* out = logm + (size_t)b * VDIM;
    for (int i = 0; i < DIM1; ++i) {
      float acc = 0.0f;
      for (int p = 0; p < DIM1; ++p) acc = fmaf(U[i][p], vj[p], acc);
      out[i * DIM1 + j] = acc;
    }
  }
}

// ---------------------------------------------------------------------------
// Kernel 3: out = logm(8192x900) * lin_w^T(900x117) + lin_b, fp32 WMMA.
// Grid: 512 M-tiles; 8 waves per block each own one 16-wide N-tile (117->128
// padded with guards). K = 900 = 225 wmma steps. lin_w (0.4 MB) lives in L2.
// ---------------------------------------------------------------------------
__global__ __launch_bounds__(256) void linear_kernel(
    const float* __restrict__ logm, const float* __restrict__ lin_w,
    const float* __restrict__ lin_b, float* __restrict__ out)
{
  const int mt   = blockIdx.x;
  const int w    = threadIdx.x >> 5;   // N tile 0..7
  const int lane = threadIdx.x & 31;
  const int half = lane >> 4;
  const int l    = lane & 15;

  const int row = mt * 16 + l;
  const int n   = w * 16 + l;
  const float* Arow = logm + (size_t)row * VDIM;
  const float* Brow = (n < CLASSES) ? (lin_w + (size_t)n * VDIM) : nullptr;

  v8f acc = {};
  for (int k4 = 0; k4 < VDIM / 4; ++k4) {
    const int kb = k4 * 4 + half * 2;
    v2f a, bb;
    a.x = Arow[kb]; a.y = Arow[kb + 1];
    if (Brow) { bb.x = Brow[kb]; bb.y = Brow[kb + 1]; }
    else      { bb.x = 0.0f;     bb.y = 0.0f;         }
    acc = __builtin_amdgcn_wmma_f32_16x16x4_f32(
        false, a, false, bb, (short)0, acc, false, false);
  }

  if (n < CLASSES) {
    const float bias = lin_b[n];
    for (int r = 0; r < 8; ++r) {
      const int rr = mt * 16 + r + half * 8;
      out[(size_t)rr * CLASSES + n] = acc[r] + bias;
    }
  }
}

// ---------------------------------------------------------------------------
extern "C" void kernel_launch(void* const* d_in, const int* in_sizes, int n_in,
                              void* d_out, int out_size, void* d_ws, size_t ws_size,
                              hipStream_t stream) {
  const float* x     = (const float*)d_in[0];  // [8192,1,93,93]
  const float* Wg    = (const float*)d_in[1];  // [93,30]
  const float* lin_w = (const float*)d_in[2];  // [117,900]
  const float* lin_b = (const float*)d_in[3];  // [117]
  float* out = (float*)d_out;                  // [8192,117]

  float* ws_y    = (float*)d_ws;                // [8192,900]
  float* ws_logm = ws_y + (size_t)BATCH * VDIM; // [8192,900]

  bimap_kernel  <<<BATCH, 256, 0, stream>>>(x, Wg, ws_y);
  eig_log_kernel<<<BATCH,  32, 0, stream>>>(ws_y, ws_logm);
  linear_kernel <<<BATCH / 16, 256, 0, stream>>>(ws_logm, lin_w, lin_b, out);
}